// TrajectoryAwareWhere2comm_24352464570102
// MI455X (gfx1250) — compile-verified
//
#include <hip/hip_runtime.h>
#include <hip/hip_bf16.h>
#include <math.h>

typedef float v2f __attribute__((ext_vector_type(2)));
typedef float v8f __attribute__((ext_vector_type(8)));

#define H4 25
#define W4 88
#define HH 100
#define WWW 352
#define HW (HH * WWW)
#define NBL 20          // B*L = 4*5
#define LAG 5
#define CCH 64
#define NB 4

__device__ __forceinline__ int iclamp(int v, int lo, int hi) {
    return v < lo ? lo : (v > hi ? hi : v);
}

// ---------------------------------------------------------------------------
// K1: per-(b,l) trajectory params: MLP (WMMA f32 GEMM), has_traj, 9 border-
// class attention scalars from the spatially-constant conv3x3 + conv1x1.
// One wave (32 threads) per block, 20 blocks.
// ---------------------------------------------------------------------------
__global__ __launch_bounds__(32)
void k_traj_params(const float* __restrict__ traj,
                   const float* __restrict__ W1, const float* __restrict__ b1,
                   const float* __restrict__ W2, const float* __restrict__ b2,
                   const float* __restrict__ Wc1, const float* __restrict__ bc1,
                   const float* __restrict__ Wc2, const float* __restrict__ bc2,
                   float* __restrict__ att9)
{
    const int n = blockIdx.x;
    const int t = threadIdx.x;
    __shared__ float ts[32];            // 10x3 trajectory points
    __shared__ float hs[16][64];        // relu(traj@W1+b1), rows 10..15 zero
    __shared__ float fmean[128];        // mean over T of h@W2 + b2
    __shared__ float P[9][64];          // per-tap matvec Wc1[:,:,ki,kj]@fmean
    __shared__ float fpart[16][2];
    __shared__ int flag;

    if (t == 0) flag = 0;
    if (t < 30) ts[t] = traj[n * 30 + t];
    __syncthreads();
    if (t < 30 && ts[t] != 0.0f) flag = 1;   // benign same-value race

    // h = relu(traj @ W1 + b1), zero-padded to 16 rows
    for (int idx = t; idx < 16 * 64; idx += 32) {
        int m = idx >> 6, j = idx & 63;
        float v = 0.0f;
        if (m < 10) {
            v = ts[m * 3 + 0] * W1[j] + ts[m * 3 + 1] * W1[64 + j]
              + ts[m * 3 + 2] * W1[128 + j] + b1[j];
            v = fmaxf(v, 0.0f);
        }
        hs[m][j] = v;
    }
    __syncthreads();

    // f = h @ W2  via V_WMMA_F32_16X16X4_F32 : M=16(10 used), K=64, N=128
    const int half = t >> 4;     // lane half selects K / M split per ISA layout
    const int lm = t & 15;
    for (int tile = 0; tile < 8; ++tile) {
        v8f c = {0.f, 0.f, 0.f, 0.f, 0.f, 0.f, 0.f, 0.f};
        for (int kk = 0; kk < 16; ++kk) {
            int ks = kk * 4 + 2 * half;
            v2f a, bb;
            a.x = hs[lm][ks];
            a.y = hs[lm][ks + 1];
            bb.x = W2[(size_t)ks * 128 + tile * 16 + lm];
            bb.y = W2[(size_t)(ks + 1) * 128 + tile * 16 + lm];
            c = __builtin_amdgcn_wmma_f32_16x16x4_f32(
                    false, a, false, bb, (short)0, c, false, false);
        }
        // sum rows M=0..9: half0 holds M=0..7 (v0..7), half1 holds M=8,9 (v0,v1)
        float p = (half == 0)
                    ? (c[0] + c[1] + c[2] + c[3] + c[4] + c[5] + c[6] + c[7])
                    : (c[0] + c[1]);
        fpart[lm][half] = p;
        __syncthreads();
        if (t < 16) {
            int col = tile * 16 + t;
            fmean[col] = (fpart[t][0] + fpart[t][1]) / 10.0f + b2[col];
        }
        __syncthreads();
    }

    // Per-tap matvecs: P[ki*3+kj][o] = sum_i Wc1[o,i,ki,kj] * fmean[i]
    for (int idx = t; idx < 9 * 64; idx += 32) {
        int kk = idx >> 6, o = idx & 63;
        int ki = kk / 3, kj = kk % 3;
        const float* wp = Wc1 + (size_t)o * 1152 + ki * 3 + kj;
        float s = 0.0f;
        for (int i = 0; i < 128; ++i) s += wp[(size_t)i * 9] * fmean[i];
        P[kk][o] = s;
    }
    __syncthreads();

    // 9 border classes -> attention scalar (folded * has_traj)
    if (t < 9) {
        int rt = t / 3, ct = t % 3;
        int ki0 = (rt == 0) ? 1 : 0, ki1 = (rt == 2) ? 1 : 2;
        int kj0 = (ct == 0) ? 1 : 0, kj1 = (ct == 2) ? 1 : 2;
        float s = 0.0f;
        for (int o = 0; o < 64; ++o) {
            float acc = bc1[o];
            for (int ki = ki0; ki <= ki1; ++ki)
                for (int kj = kj0; kj <= kj1; ++kj)
                    acc += P[ki * 3 + kj][o];
            acc = fmaxf(acc, 0.0f);
            s += Wc2[o] * acc;
        }
        float att = 1.0f / (1.0f + expf(-(s + bc2[0])));
        att9[n * 9 + t] = att * (flag ? 1.0f : 0.0f);
    }
}

// ---------------------------------------------------------------------------
// K2: gaussian splat of trajectory points at H4xW4, L1-normalize, * att class
// ---------------------------------------------------------------------------
__global__ __launch_bounds__(256)
void k_splat(const float* __restrict__ traj, const float* __restrict__ att9,
             float* __restrict__ tm4)
{
    const int n = blockIdx.x;
    const int t = threadIdx.x;
    __shared__ int xi[10], yi[10];
    __shared__ float buf[H4 * W4];
    __shared__ float red[256];

    if (t < 10) {
        float px = traj[n * 30 + t * 3 + 0] / 0.4f;   // / DISCRETE_RATIO
        float py = traj[n * 30 + t * 3 + 1] / 0.4f;
        xi[t] = iclamp((int)(px * 0.25f), 0, W4 - 1); // /4 then trunc
        yi[t] = iclamp((int)(py * 0.25f), 0, H4 - 1);
    }
    __syncthreads();

    float local = 0.0f;
    for (int p = t; p < H4 * W4; p += 256) {
        int i = p / W4, j = p % W4;
        float m = 0.0f;
        for (int tt = 0; tt < 10; ++tt) {
            int di = i - yi[tt], dj = j - xi[tt];
            if (di >= -3 && di <= 3 && dj >= -3 && dj <= 3)
                m += expf(-(float)(di * di + dj * dj) * 0.125f); // 2*sigma^2=8
        }
        buf[p] = m;
        local += m;
    }
    red[t] = local;
    __syncthreads();
    for (int s = 128; s > 0; s >>= 1) {
        if (t < s) red[t] += red[t + s];
        __syncthreads();
    }
    float inv = 1.0f / fmaxf(red[0], 1e-12f);
    for (int p = t; p < H4 * W4; p += 256) {
        int i = p / W4, j = p % W4;
        int rt = (i == 0) ? 0 : ((i == H4 - 1) ? 2 : 1);
        int ct = (j == 0) ? 0 : ((j == W4 - 1) ? 2 : 1);
        tm4[(size_t)n * (H4 * W4) + p] = buf[p] * inv * att9[n * 9 + rt * 3 + ct];
    }
}

// ---------------------------------------------------------------------------
// K3: clamped half-pixel bilinear x4 upsample + max_a sigmoid(psm*(1+tm))
// ---------------------------------------------------------------------------
__global__ __launch_bounds__(256)
void k_cmax(const float* __restrict__ psm, const float* __restrict__ tm4,
            float* __restrict__ cmax)
{
    int idx = blockIdx.x * blockDim.x + threadIdx.x;
    if (idx >= NBL * HW) return;
    int n = idx / HW, p = idx % HW;
    int h = p / WWW, w = p % WWW;

    float fy = ((float)h + 0.5f) * 0.25f - 0.5f;
    float fx = ((float)w + 0.5f) * 0.25f - 0.5f;
    int y0 = (int)floorf(fy), x0 = (int)floorf(fx);
    float wy = fy - (float)y0, wx = fx - (float)x0;
    int y0c = iclamp(y0, 0, H4 - 1), y1c = iclamp(y0 + 1, 0, H4 - 1);
    int x0c = iclamp(x0, 0, W4 - 1), x1c = iclamp(x0 + 1, 0, W4 - 1);
    const float* tb = tm4 + (size_t)n * (H4 * W4);
    float v00 = tb[y0c * W4 + x0c], v01 = tb[y0c * W4 + x1c];
    float v10 = tb[y1c * W4 + x0c], v11 = tb[y1c * W4 + x1c];
    float tm = (1.0f - wy) * ((1.0f - wx) * v00 + wx * v01)
             + wy * ((1.0f - wx) * v10 + wx * v11);

    float f = 1.0f + tm;
    float a0 = psm[((size_t)n * 2 + 0) * HW + p] * f;
    float a1 = psm[((size_t)n * 2 + 1) * HW + p] * f;
    float s0 = 1.0f / (1.0f + expf(-a0));
    float s1 = 1.0f / (1.0f + expf(-a1));
    cmax[idx] = fmaxf(s0, s1);
}

// ---------------------------------------------------------------------------
// K4: fixed 5x5 gaussian smooth (zero pad), threshold, ego mask forced to 1
// ---------------------------------------------------------------------------
__global__ __launch_bounds__(256)
void k_mask(const float* __restrict__ cmax, float* __restrict__ mask)
{
    int idx = blockIdx.x * blockDim.x + threadIdx.x;
    if (idx >= NBL * HW) return;
    int n = idx / HW, p = idx % HW;
    int h = p / WWW, w = p % WWW;
    float s = 0.0f;
    for (int dy = -2; dy <= 2; ++dy) {
        int hh = h + dy;
        if (hh < 0 || hh >= HH) continue;
        for (int dx = -2; dx <= 2; ++dx) {
            int ww = w + dx;
            if (ww < 0 || ww >= WWW) continue;
            float g = 0.15915494309189535f * expf(-(float)(dy * dy + dx * dx) * 0.5f);
            s += g * cmax[(size_t)n * HW + hh * WWW + ww];
        }
    }
    float m = (s > 0.01f) ? 1.0f : 0.0f;
    if ((n % LAG) == 0) m = 1.0f;   // ego forced
    mask[idx] = m;
}

// ---------------------------------------------------------------------------
// K5: per-pixel ego-row attention over L=5 agents, C=64
// ---------------------------------------------------------------------------
__global__ __launch_bounds__(256)
void k_fuse(const float* __restrict__ x, const float* __restrict__ mask,
            float* __restrict__ out)
{
    int idx = blockIdx.x * blockDim.x + threadIdx.x;
    if (idx >= NB * HW) return;
    int b = idx / HW, p = idx % HW;
    const float* xb = x + (size_t)b * LAG * CCH * HW + p;

    float x0[64];
    #pragma unroll
    for (int c = 0; c < 64; ++c) x0[c] = xb[(size_t)c * HW];

    float sc[5], mm[5];
    float d0 = 0.0f;
    #pragma unroll
    for (int c = 0; c < 64; ++c) d0 += x0[c] * x0[c];
    sc[0] = d0 * 0.125f;    // scale = 1/sqrt(64)
    mm[0] = 1.0f;
    for (int m = 1; m < 5; ++m) {
        mm[m] = mask[((size_t)b * LAG + m) * HW + p];
        const float* xm = xb + (size_t)m * CCH * HW;
        float d = 0.0f;
        #pragma unroll
        for (int c = 0; c < 64; ++c) d += x0[c] * xm[(size_t)c * HW];
        sc[m] = d * mm[m] * 0.125f;     // v_m already masked
    }
    float mx = sc[0];
    for (int m = 1; m < 5; ++m) mx = fmaxf(mx, sc[m]);
    float w5[5], den = 0.0f;
    for (int m = 0; m < 5; ++m) { w5[m] = expf(sc[m] - mx); den += w5[m]; }
    float invd = 1.0f / den;

    float acc[64];
    #pragma unroll
    for (int c = 0; c < 64; ++c) acc[c] = (w5[0] * invd) * x0[c];
    for (int m = 1; m < 5; ++m) {
        float wm = w5[m] * invd * mm[m];
        const float* xm = xb + (size_t)m * CCH * HW;
        #pragma unroll
        for (int c = 0; c < 64; ++c) acc[c] += wm * xm[(size_t)c * HW];
    }
    float* ob = out + (size_t)b * CCH * HW + p;
    #pragma unroll
    for (int c = 0; c < 64; ++c) ob[(size_t)c * HW] = acc[c];
}

// ---------------------------------------------------------------------------
extern "C" void kernel_launch(void* const* d_in, const int* in_sizes, int n_in,
                              void* d_out, int out_size, void* d_ws, size_t ws_size,
                              hipStream_t stream)
{
    const float* x    = (const float*)d_in[0];
    const float* psm  = (const float*)d_in[1];
    // d_in[2] record_len, d_in[3] pairwise_t_matrix: unused by reference path
    const float* traj = (const float*)d_in[4];
    const float* W1   = (const float*)d_in[5];
    const float* b1   = (const float*)d_in[6];
    const float* W2   = (const float*)d_in[7];
    const float* b2   = (const float*)d_in[8];
    const float* Wc1  = (const float*)d_in[9];
    const float* bc1  = (const float*)d_in[10];
    const float* Wc2  = (const float*)d_in[11];
    const float* bc2  = (const float*)d_in[12];
    float* out = (float*)d_out;

    float* ws   = (float*)d_ws;
    float* att9 = ws;              // 20*9   = 180   floats
    float* tm4  = ws + 256;        // 20*2200 = 44000 floats
    float* cmax = ws + 45056;      // 20*35200 = 704000 floats
    float* mask = ws + 749568;     // 20*35200 = 704000 floats (~5.8 MB total)

    k_traj_params<<<NBL, 32, 0, stream>>>(traj, W1, b1, W2, b2,
                                          Wc1, bc1, Wc2, bc2, att9);
    k_splat<<<NBL, 256, 0, stream>>>(traj, att9, tm4);
    int tot = NBL * HW;
    k_cmax<<<(tot + 255) / 256, 256, 0, stream>>>(psm, tm4, cmax);
    k_mask<<<(tot + 255) / 256, 256, 0, stream>>>(cmax, mask);
    k_fuse<<<(NB * HW + 255) / 256, 256, 0, stream>>>(x, mask, out);
}